// CTRGC_81655918232207
// MI455X (gfx1250) — compile-verified
//
#include <hip/hip_runtime.h>

// CTR-GCN block for MI455X (gfx1250).
// fp32 WMMA (V_WMMA_F32_16X16X4_F32) + TDM tensor_load_to_lds staging.
// N=128, C=64, T=128, V=25, R=8, O=64.

#define ALPHA_F 1.0f

constexpr int N_ = 128, C_ = 64, T_ = 128, V_ = 25, R_ = 8, O_ = 64;
constexpr int TT = 16;           // t-tile per block
constexpr int TV = TT * V_;      // 400 flattened (t,v) per tile

typedef float v2f __attribute__((ext_vector_type(2)));
typedef float v8f __attribute__((ext_vector_type(8)));
typedef unsigned int u32x4 __attribute__((ext_vector_type(4)));
typedef int i32x4 __attribute__((ext_vector_type(4)));
typedef int i32x8 __attribute__((ext_vector_type(8)));

// ---------------------------------------------------------------- kernel 1
// xm[n,c,v] = mean_t x[n,c,t,v]
__global__ __launch_bounds__(256) void k_mean(const float* __restrict__ x,
                                              float* __restrict__ xm) {
    int idx = blockIdx.x * 256 + threadIdx.x;
    if (idx >= N_ * C_ * V_) return;
    int v  = idx % V_;
    int nc = idx / V_;
    const float* p = x + (size_t)nc * T_ * V_ + v;
    float s = 0.0f;
#pragma unroll 8
    for (int t = 0; t < T_; ++t) s += p[(size_t)t * V_];
    xm[idx] = s * (1.0f / (float)T_);
}

// ---------------------------------------------------------------- kernel 2
// top[n,o,u,v] = alpha*(sum_r w4[o,r]*tanh(x1[n,r,u]-x2[n,r,v]) + b4[o]) + A[u,v]
__global__ __launch_bounds__(256) void k_top(
    const float* __restrict__ xm, const float* __restrict__ A,
    const float* __restrict__ w1, const float* __restrict__ b1,
    const float* __restrict__ w2, const float* __restrict__ b2,
    const float* __restrict__ w4, const float* __restrict__ b4,
    float* __restrict__ top) {
    __shared__ float xms[C_ * V_];          // 1600 f
    __shared__ float x1s[R_ * V_];          // 200 f
    __shared__ float x2s[R_ * V_];          // 200 f
    __shared__ float ds[R_ * V_ * V_];      // 5000 f

    const int n   = blockIdx.x;
    const int tid = threadIdx.x;

    for (int i = tid; i < C_ * V_; i += 256)
        xms[i] = xm[(size_t)n * C_ * V_ + i];
    __syncthreads();

    for (int i = tid; i < R_ * V_; i += 256) {
        int r = i / V_, v = i - r * V_;
        float s1 = b1[r], s2 = b2[r];
        for (int c = 0; c < C_; ++c) {
            float xv = xms[c * V_ + v];
            s1 += xv * w1[r * C_ + c];
            s2 += xv * w2[r * C_ + c];
        }
        x1s[i] = s1;
        x2s[i] = s2;
    }
    __syncthreads();

    for (int i = tid; i < R_ * V_ * V_; i += 256) {
        int r  = i / (V_ * V_);
        int uv = i - r * (V_ * V_);
        int u  = uv / V_, v = uv - u * V_;
        ds[i] = tanhf(x1s[r * V_ + u] - x2s[r * V_ + v]);
    }
    __syncthreads();

    float* topn = top + (size_t)n * O_ * V_ * V_;
    for (int i = tid; i < O_ * V_ * V_; i += 256) {
        int o  = i / (V_ * V_);
        int uv = i - o * (V_ * V_);
        float s = b4[o];
#pragma unroll
        for (int r = 0; r < R_; ++r) s += w4[o * R_ + r] * ds[r * V_ * V_ + uv];
        topn[i] = s * ALPHA_F + A[uv];
    }
}

// ---------------------------------------------------------------- kernel 3
// Fused:  x3[n,o,t,v] = sum_c w3[o,c] x[n,c,t,v] + b3[o]   (GEMM1, via LDS)
//         out[n,o,t,u] = sum_v top[n,o,u,v] * x3[n,o,t,v]  (GEMM2)
// Block = (n, 16-row t tile), 8 waves. TDM stages the 100 KB x-tile into LDS.
__global__ __launch_bounds__(256) void k_gemm(
    const float* __restrict__ x, const float* __restrict__ w3,
    const float* __restrict__ b3, const float* __restrict__ top,
    float* __restrict__ out) {
    __shared__ float Xs[C_ * TV];    // [c][t*25+v]  : 102400 B
    __shared__ float X3s[16 * TV];   // [o_loc][t*25+v] : 25600 B
    __shared__ float W3s[O_ * C_];   // 16384 B
    __shared__ float b3s[O_];

    const int tid  = threadIdx.x;
    const int wave = tid >> 5;
    const int lane = tid & 31;
    const int half = lane >> 4;   // K-half selector (ISA f32 A/B layout)
    const int mn   = lane & 15;   // M (A) / N (B,C,D) index

    const int n   = blockIdx.x >> 3;          // T_/TT == 8 tiles
    const int tt0 = (blockIdx.x & 7) * TT;

    // ---- stage x[n, :, tt0:tt0+16, :] -> Xs via Tensor Data Mover:
    //      2D tile: 64 rows (c) x 400 contiguous floats, row stride T*V elems.
    const float* xn = x + ((size_t)n * C_ * T_ + tt0) * V_;
#if __has_builtin(__builtin_amdgcn_tensor_load_to_lds)
    if (tid < 32) {   // wave 0 issues the DMA (EXEC ignored by TDM)
        unsigned long long ga = (unsigned long long)xn;
        unsigned lds_base =
            (unsigned)(unsigned long long)(__attribute__((address_space(3))) float*)Xs;
        u32x4 g0;
        g0[0] = 1u;                                   // count=1, user mode
        g0[1] = lds_base;                             // lds_addr (bytes)
        g0[2] = (unsigned)ga;                         // global_addr[31:0]
        g0[3] = (unsigned)((ga >> 32) & 0x1FFFFFFull) // global_addr[56:32]
                | 0x80000000u;                        // type=2 ("image")
        i32x8 g1;
        g1[0] = (int)(2u << 16);                      // data_size=4B; mask=0
        g1[1] = (int)((unsigned)(TV & 0xFFFF) << 16); // tensor_dim0 lo16
        g1[2] = (int)(((TV >> 16) & 0xFFFF)           // tensor_dim0 hi16
                | ((unsigned)(C_ & 0xFFFF) << 16));   // tensor_dim1 lo16
        g1[3] = (int)(((C_ >> 16) & 0xFFFF)           // tensor_dim1 hi16
                | ((unsigned)(TV & 0xFFFF) << 16));   // tile_dim0 = 400
        g1[4] = (int)(C_ & 0xFFFF);                   // tile_dim1 = 64, tile_dim2=0
        g1[5] = (int)(T_ * V_);                       // tensor_dim0_stride = 3200
        g1[6] = 0;                                    // stride0 hi / stride1 lo
        g1[7] = 0;                                    // stride1 hi
        i32x4 gz4 = {0, 0, 0, 0};                     // groups 2/3 unused (2D)
        i32x8 gz8 = {0, 0, 0, 0, 0, 0, 0, 0};
        __builtin_amdgcn_tensor_load_to_lds(g0, g1, gz4, gz4, gz8, 0);
    }
#else
    for (int i = tid; i < C_ * TV; i += 256) {
        int c = i / TV, s = i - c * TV;
        Xs[i] = xn[(size_t)c * T_ * V_ + s];
    }
#endif
    // meanwhile all waves stage the (small) weights
    for (int i = tid; i < O_ * C_; i += 256) W3s[i] = w3[i];
    if (tid < O_) b3s[tid] = b3[tid];
#if __has_builtin(__builtin_amdgcn_tensor_load_to_lds)
    if (tid < 32) __builtin_amdgcn_s_wait_tensorcnt(0);
#endif
    __syncthreads();

    for (int ot = 0; ot < O_ / 16; ++ot) {
        const int obase = ot * 16;

        // ---- GEMM1: X3s[o_loc, tv] = sum_c W3s[obase+o_loc, c] * Xs[c, tv]
        //      M=o(16) x K=C(64) x N=tv(400 -> 25 tiles of 16)
        for (int nt = wave; nt < TV / 16; nt += 8) {
            const int tvb = nt * 16;
            v8f acc = {};
#pragma unroll
            for (int kb = 0; kb < C_; kb += 4) {
                const int k0 = kb + half * 2;
                v2f a, b;
                a[0] = W3s[(obase + mn) * C_ + k0];
                a[1] = W3s[(obase + mn) * C_ + k0 + 1];
                b[0] = Xs[(k0)     * TV + tvb + mn];
                b[1] = Xs[(k0 + 1) * TV + tvb + mn];
                acc = __builtin_amdgcn_wmma_f32_16x16x4_f32(
                    false, a, false, b, (short)0, acc, false, false);
            }
#pragma unroll
            for (int r = 0; r < 8; ++r) {
                int ol = half * 8 + r;                    // D row = M = o_loc
                X3s[ol * TV + tvb + mn] = acc[r] + b3s[obase + ol];
            }
        }
        __syncthreads();

        // ---- GEMM2 per o: OUT[t(16) x u(25->32)] = X3[t,v] * top[u,v]^T
        //      K = v = 25 padded to 28 (7 k-steps of 4)
        for (int p = wave; p < 32; p += 8) {
            const int ol = p >> 1, ntile = p & 1;
            const int o  = obase + ol;
            const float* topo = top + ((size_t)n * O_ + o) * V_ * V_; // [u][v]
            const int ucol = ntile * 16 + mn;
            v8f acc = {};
#pragma unroll
            for (int kb = 0; kb < 28; kb += 4) {
                const int k0 = kb + half * 2;
                const int k1 = k0 + 1;
                v2f a, b;
                a[0] = (k0 < V_) ? X3s[ol * TV + mn * V_ + k0] : 0.0f;
                a[1] = (k1 < V_) ? X3s[ol * TV + mn * V_ + k1] : 0.0f;
                b[0] = (k0 < V_ && ucol < V_) ? topo[ucol * V_ + k0] : 0.0f;
                b[1] = (k1 < V_ && ucol < V_) ? topo[ucol * V_ + k1] : 0.0f;
                acc = __builtin_amdgcn_wmma_f32_16x16x4_f32(
                    false, a, false, b, (short)0, acc, false, false);
            }
            if (ucol < V_) {
                float* po = out +
                    (((size_t)n * O_ + o) * T_ + tt0 + half * 8) * V_ + ucol;
#pragma unroll
                for (int r = 0; r < 8; ++r) po[(size_t)r * V_] = acc[r];
            }
        }
        __syncthreads();
    }
}

// ---------------------------------------------------------------- launcher
extern "C" void kernel_launch(void* const* d_in, const int* in_sizes, int n_in,
                              void* d_out, int out_size, void* d_ws, size_t ws_size,
                              hipStream_t stream) {
    (void)in_sizes; (void)n_in; (void)out_size; (void)ws_size;
    const float* x  = (const float*)d_in[0];
    const float* A  = (const float*)d_in[1];
    const float* w1 = (const float*)d_in[2];
    const float* b1 = (const float*)d_in[3];
    const float* w2 = (const float*)d_in[4];
    const float* b2 = (const float*)d_in[5];
    const float* w3 = (const float*)d_in[6];
    const float* b3 = (const float*)d_in[7];
    const float* w4 = (const float*)d_in[8];
    const float* b4 = (const float*)d_in[9];
    float* out = (float*)d_out;

    float* xm  = (float*)d_ws;                          // N*C*V   = 204,800 f
    float* top = xm + (size_t)N_ * C_ * V_;             // N*O*V*V = 5,120,000 f

    k_mean<<<(N_ * C_ * V_ + 255) / 256, 256, 0, stream>>>(x, xm);
    k_top<<<N_, 256, 0, stream>>>(xm, A, w1, b1, w2, b2, w4, b4, top);
    k_gemm<<<N_ * (T_ / TT), 256, 0, stream>>>(x, w3, b3, top, out);
}